// BumpKNN_57397942944389
// MI455X (gfx1250) — compile-verified
//
#include <hip/hip_runtime.h>
#include <math.h>

typedef __attribute__((ext_vector_type(2))) float v2f;
typedef __attribute__((ext_vector_type(8))) float v8f;

#define NQ      2048
#define ND      65536
#define DIM     16
#define KAUG    20              // augmented K: 16 dims + d^2 slot + 3 pad (5 chunks of 4)
#define NSPLIT  32
#define QT      (NQ / 16)       // 128 query tiles of 16
#define CHUNK   (ND / NSPLIT)   // 2048 data rows per split
#define NTILES  (CHUNK / 16)    // 128 data tiles of 16 per wave
#define TSTRIDE ((16 * KAUG) / 2)  // float2 stride between 16-row tiles (=160)

// Raw v_min_num_f32: skips the fminf NaN-canonicalize (v_max v,v,v) dance.
// Inputs here are finite by construction.
__device__ __forceinline__ float fmin_raw(float a, float b) {
  float d;
  asm("v_min_num_f32 %0, %1, %2" : "=v"(d) : "v"(a), "v"(b));
  return d;
}

// ---------------------------------------------------------------------------
// Prep: B'[n] = [-2*data[n][0..15], ||data[n]||^2, 0, 0, 0]  (rows of 20 f32)
// ---------------------------------------------------------------------------
__global__ __launch_bounds__(256)
void knn_prep(const float* __restrict__ data, float* __restrict__ bp) {
  int n = blockIdx.x * blockDim.x + threadIdx.x;
  if (n >= ND) return;
  const float4* src = (const float4*)(data + (size_t)n * DIM);
  float4 v0 = src[0], v1 = src[1], v2 = src[2], v3 = src[3];
  float d2 = v0.x * v0.x + v0.y * v0.y + v0.z * v0.z + v0.w * v0.w +
             v1.x * v1.x + v1.y * v1.y + v1.z * v1.z + v1.w * v1.w +
             v2.x * v2.x + v2.y * v2.y + v2.z * v2.z + v2.w * v2.w +
             v3.x * v3.x + v3.y * v3.y + v3.z * v3.z + v3.w * v3.w;
  float4* dst = (float4*)(bp + (size_t)n * KAUG);   // 80B rows, 16B aligned
  dst[0] = make_float4(-2.f * v0.x, -2.f * v0.y, -2.f * v0.z, -2.f * v0.w);
  dst[1] = make_float4(-2.f * v1.x, -2.f * v1.y, -2.f * v1.z, -2.f * v1.w);
  dst[2] = make_float4(-2.f * v2.x, -2.f * v2.y, -2.f * v2.z, -2.f * v2.w);
  dst[3] = make_float4(-2.f * v3.x, -2.f * v3.y, -2.f * v3.z, -2.f * v3.w);
  dst[4] = make_float4(d2, 0.f, 0.f, 0.f);
}

// ---------------------------------------------------------------------------
// Main: one wave per (query-tile, data-split). acc = d^2 - 2 x.d via 5 chained
// V_WMMA_F32_16X16X4_F32; double-buffered B-tile loads (prefetch distance 2);
// inner-loop epilogue is raw v_min only; x^2 added at the partial write.
// ---------------------------------------------------------------------------
__global__ __launch_bounds__(32)
void knn_wmma_main(const float* __restrict__ x,
                   const float* __restrict__ bp,
                   float* __restrict__ partial) {
  const int qtile = blockIdx.x;   // 0..QT-1
  const int split = blockIdx.y;   // 0..NSPLIT-1
  const int lane  = threadIdx.x;  // 0..31 (wave32)
  const int col   = lane & 15;
  const int half  = lane >> 4;

  // A' tile: 16 queries x 20 in WMMA f32 16x16x4 A layout.
  // Chunk kc holds K = 4kc..4kc+3; lane(row=col, khalf=half) holds elements
  // {4kc+2h, 4kc+2h+1}. Chunk 4 is the constant [1,0,0,0] augmentation.
  const float2* xrow = (const float2*)(x + (size_t)(qtile * 16 + col) * DIM);
  v2f a[5];
  float px2 = 0.f;
#pragma unroll
  for (int kc = 0; kc < 4; ++kc) {
    float2 t = xrow[2 * kc + half];
    a[kc].x = t.x;
    a[kc].y = t.y;
    px2 += t.x * t.x + t.y * t.y;
  }
  a[4].x = (half == 0) ? 1.0f : 0.0f;   // elements {16,17} or {18,19}
  a[4].y = 0.0f;

  // ||x_m||^2 for m = col (combine K-halves), then gather the 8 rows this
  // lane owns in the C layout (M = r + 8*half). One-time, outside the loop.
  float x2full = px2 + __shfl_xor(px2, 16, 32);
  float x2v[8];
#pragma unroll
  for (int r = 0; r < 8; ++r)
    x2v[r] = __shfl(x2full, r + 8 * half, 32);

  float minv[8];
#pragma unroll
  for (int r = 0; r < 8; ++r) minv[r] = 3.4e38f;

  const float2* p =
      (const float2*)(bp + ((size_t)split * CHUNK + col) * KAUG);

  // Prologue: fill both B-tile buffers.
  v2f b0[5], b1[5];
#pragma unroll
  for (int kc = 0; kc < 5; ++kc) {
    float2 u = p[2 * kc + half];
    b0[kc].x = u.x; b0[kc].y = u.y;
  }
  p += TSTRIDE;
#pragma unroll
  for (int kc = 0; kc < 5; ++kc) {
    float2 u = p[2 * kc + half];
    b1[kc].x = u.x; b1[kc].y = u.y;
  }
  p += TSTRIDE;

  // Steady state: compute tile t from b0/b1 while loading tile t+2 into it.
  for (int t = 0; t < NTILES - 2; t += 2) {
    v8f acc0 = {};
#pragma unroll
    for (int kc = 0; kc < 5; ++kc)
      acc0 = __builtin_amdgcn_wmma_f32_16x16x4_f32(
          false, a[kc], false, b0[kc], (short)0, acc0, false, false);
#pragma unroll
    for (int kc = 0; kc < 5; ++kc) {      // refill b0 with tile t+2
      float2 u = p[2 * kc + half];
      b0[kc].x = u.x; b0[kc].y = u.y;
    }
    p += TSTRIDE;
#pragma unroll
    for (int r = 0; r < 8; ++r)
      minv[r] = fmin_raw(minv[r], acc0[r]);

    v8f acc1 = {};
#pragma unroll
    for (int kc = 0; kc < 5; ++kc)
      acc1 = __builtin_amdgcn_wmma_f32_16x16x4_f32(
          false, a[kc], false, b1[kc], (short)0, acc1, false, false);
#pragma unroll
    for (int kc = 0; kc < 5; ++kc) {      // refill b1 with tile t+3
      float2 u = p[2 * kc + half];
      b1[kc].x = u.x; b1[kc].y = u.y;
    }
    p += TSTRIDE;
#pragma unroll
    for (int r = 0; r < 8; ++r)
      minv[r] = fmin_raw(minv[r], acc1[r]);
  }

  // Epilogue: last two tiles (already resident in b0/b1).
  {
    v8f acc0 = {};
#pragma unroll
    for (int kc = 0; kc < 5; ++kc)
      acc0 = __builtin_amdgcn_wmma_f32_16x16x4_f32(
          false, a[kc], false, b0[kc], (short)0, acc0, false, false);
#pragma unroll
    for (int r = 0; r < 8; ++r)
      minv[r] = fmin_raw(minv[r], acc0[r]);

    v8f acc1 = {};
#pragma unroll
    for (int kc = 0; kc < 5; ++kc)
      acc1 = __builtin_amdgcn_wmma_f32_16x16x4_f32(
          false, a[kc], false, b1[kc], (short)0, acc1, false, false);
#pragma unroll
    for (int r = 0; r < 8; ++r)
      minv[r] = fmin_raw(minv[r], acc1[r]);
  }

  // Min over the 16 columns: butterfly within each 16-lane half-group
#pragma unroll
  for (int s = 1; s < 16; s <<= 1) {
#pragma unroll
    for (int r = 0; r < 8; ++r)
      minv[r] = fmin_raw(minv[r], __shfl_xor(minv[r], s, 32));
  }

  if (col == 0) {
#pragma unroll
    for (int r = 0; r < 8; ++r) {
      int m = r + 8 * half;   // query row within the tile
      partial[(size_t)(qtile * 16 + m) * NSPLIT + split] = minv[r] + x2v[r];
    }
  }
}

// ---------------------------------------------------------------------------
// Fallback (self-contained, no big scratch): d^2 computed in-loop; -2 folded
// into A so the per-tile term is d^2 + acc; x^2 added at the write.
// ---------------------------------------------------------------------------
__global__ __launch_bounds__(32)
void knn_wmma_fallback(const float* __restrict__ x,
                       const float* __restrict__ data,
                       float* __restrict__ partial) {
  const int qtile = blockIdx.x;
  const int split = blockIdx.y;
  const int lane  = threadIdx.x;
  const int col   = lane & 15;
  const int half  = lane >> 4;

  const float2* xrow = (const float2*)(x + (size_t)(qtile * 16 + col) * DIM);
  v2f a[4];
  float px2 = 0.f;
#pragma unroll
  for (int kc = 0; kc < 4; ++kc) {
    float2 t = xrow[2 * kc + half];
    a[kc].x = -2.0f * t.x;
    a[kc].y = -2.0f * t.y;
    px2 += t.x * t.x + t.y * t.y;
  }
  float x2full = px2 + __shfl_xor(px2, 16, 32);
  float x2v[8];
#pragma unroll
  for (int r = 0; r < 8; ++r)
    x2v[r] = __shfl(x2full, r + 8 * half, 32);

  float minv[8];
#pragma unroll
  for (int r = 0; r < 8; ++r) minv[r] = 3.4e38f;

  const float2* p =
      (const float2*)(data + ((size_t)split * CHUNK + col) * DIM);
  for (int t = 0; t < NTILES; ++t) {
    v2f b[4];
    float pd2 = 0.f;
#pragma unroll
    for (int kc = 0; kc < 4; ++kc) {
      float2 u = p[2 * kc + half];
      b[kc].x = u.x;
      b[kc].y = u.y;
      pd2 += u.x * u.x + u.y * u.y;
    }
    p += (16 * DIM) / 2;
    float d2 = pd2 + __shfl_xor(pd2, 16, 32);

    v8f acc = {};
#pragma unroll
    for (int kc = 0; kc < 4; ++kc)
      acc = __builtin_amdgcn_wmma_f32_16x16x4_f32(
          false, a[kc], false, b[kc], (short)0, acc, false, false);

#pragma unroll
    for (int r = 0; r < 8; ++r)
      minv[r] = fmin_raw(minv[r], d2 + acc[r]);   // d^2 - 2 x.d
  }

#pragma unroll
  for (int s = 1; s < 16; s <<= 1) {
#pragma unroll
    for (int r = 0; r < 8; ++r)
      minv[r] = fmin_raw(minv[r], __shfl_xor(minv[r], s, 32));
  }

  if (col == 0) {
#pragma unroll
    for (int r = 0; r < 8; ++r) {
      int m = r + 8 * half;
      partial[(size_t)(qtile * 16 + m) * NSPLIT + split] = minv[r] + x2v[r];
    }
  }
}

// ---------------------------------------------------------------------------
// Finalize: min across splits + clamp + sqrt + bump.
// ---------------------------------------------------------------------------
__global__ __launch_bounds__(256)
void knn_finalize(const float* __restrict__ partial, float* __restrict__ out) {
  int q = blockIdx.x * blockDim.x + threadIdx.x;
  if (q >= NQ) return;
  float m = 3.4e38f;
#pragma unroll
  for (int s = 0; s < NSPLIT; ++s)
    m = fmin_raw(m, partial[(size_t)q * NSPLIT + s]);
  float nn2  = fmaxf(m, 0.0f);
  float d2c  = fmaxf(nn2, 1e-12f);   // dist*dist, dist = sqrt(max(nn2,1e-12))
  float dist = sqrtf(d2c);
  float bump = 0.0f;
  if (dist < 2.0f)                   // RADIUS = 2, DECAY = 1, DECAY/r2 = 0.25
    bump = expf(1.0f / (d2c - 4.0f) + 0.25f);
  out[q] = bump;
}

extern "C" void kernel_launch(void* const* d_in, const int* in_sizes, int n_in,
                              void* d_out, int out_size, void* d_ws, size_t ws_size,
                              hipStream_t stream) {
  const float* x    = (const float*)d_in[0];   // [2048, 16]  f32
  const float* data = (const float*)d_in[1];   // [65536, 16] f32
  float* out        = (float*)d_out;           // [2048]      f32

  float* partial = (float*)d_ws;               // NQ*NSPLIT f32 = 256 KB
  const size_t partial_bytes = (size_t)NQ * NSPLIT * sizeof(float);
  const size_t bp_bytes      = (size_t)ND * KAUG * sizeof(float);   // 5.25 MB

  dim3 grid1(QT, NSPLIT);
  if (ws_size >= partial_bytes + bp_bytes) {
    float* bp = partial + (size_t)NQ * NSPLIT;   // 16B-aligned (256 KB offset)
    knn_prep<<<ND / 256, 256, 0, stream>>>(data, bp);
    knn_wmma_main<<<grid1, 32, 0, stream>>>(x, bp, partial);
  } else {
    knn_wmma_fallback<<<grid1, 32, 0, stream>>>(x, data, partial);
  }
  knn_finalize<<<(NQ + 255) / 256, 256, 0, stream>>>(partial, out);
}